// GraphConvolution_18665927868924
// MI455X (gfx1250) — compile-verified
//
#include <hip/hip_runtime.h>
#include <hip/hip_bf16.h>

typedef float v2f  __attribute__((ext_vector_type(2)));
typedef float v4f  __attribute__((ext_vector_type(4)));
typedef float v8f  __attribute__((ext_vector_type(8)));

#define D_FEAT 128          // D_IN == D_OUT == 128
#define LDS_STRIDE 132      // 128 + 4 pad: row stride 528B = 33*16B (keeps 16B align, kills bank conflicts)

// ---------------------------------------------------------------------------
// Device-scope non-returning fp32 atomic add (native global_atomic_add_f32).
// Inline asm guarantees the hardware atomic (no CAS expansion) and DEV scope
// (default scope 0 = WGP would not be coherent across workgroups).
// ---------------------------------------------------------------------------
__device__ __forceinline__ void global_fadd_dev(float* p, float v) {
    asm volatile("global_atomic_add_f32 %0, %1, off scope:SCOPE_DEV"
                 :: "v"(p), "v"(v)
                 : "memory");
}

// ---------------------------------------------------------------------------
// Kernel 1: support = X @ W  (exact fp32 via V_WMMA_F32_16X16X4_F32)
// Block = 256 threads = 8 waves. Block owns a 16-row M-slab; wave w owns the
// 16-column N-tile n0 = 16*w (8 tiles cover D_OUT=128).
// A-tile (16x128 fp32) staged in LDS once per block and reused by all 8 waves.
//
// Fragment layouts per CDNA5 ISA 7.12.2:
//  A 16x4 f32 : lanes 0-15 -> M=lane,  VGPR0=K0,VGPR1=K1 ; lanes 16-31 -> K2,K3
//  B 4x16 f32 : VGPR0: lanes0-15 = (K0, N=lane), lanes16-31 = (K2, N=lane-16);
//               VGPR1: K1 / K3
//  C/D 16x16  : VGPR r -> row M = r + 8*(lane>>4), col N = lane&15
// ---------------------------------------------------------------------------
__global__ void __launch_bounds__(256, 2)
gcn_gemm_f32_wmma(const float* __restrict__ X, const float* __restrict__ W,
                  float* __restrict__ support, int n_nodes) {
    __shared__ float ldsX[16 * LDS_STRIDE];

    const int lane  = threadIdx.x & 31;
    const int wave  = threadIdx.x >> 5;
    const int mbase = blockIdx.x * 16;

    // Cooperative load of the 16x128 X tile: 2048 floats, 256 threads, 8 each.
    {
        const int t    = threadIdx.x;
        const int row  = t >> 4;          // 0..15
        const int col  = (t & 15) * 8;    // 0,8,...,120
        int srow = mbase + row;
        if (srow >= n_nodes) srow = n_nodes - 1;   // clamp (tail safety)
        const v4f* src = (const v4f*)(X + (size_t)srow * D_FEAT + col);
        v4f a0 = src[0];
        v4f a1 = src[1];
        *(v4f*)(&ldsX[row * LDS_STRIDE + col + 0]) = a0;
        *(v4f*)(&ldsX[row * LDS_STRIDE + col + 4]) = a1;
    }
    __syncthreads();

    const int n0    = wave * 16;
    const int mrow  = lane & 15;           // A: M index / B: N index
    const int khalf = (lane >> 4) * 2;     // 0 for lanes 0-15, 2 for lanes 16-31

    v8f acc = {};
    #pragma unroll 4
    for (int k0 = 0; k0 < D_FEAT; k0 += 4) {
        v2f afrag, bfrag;
        afrag.x = ldsX[mrow * LDS_STRIDE + k0 + khalf + 0];
        afrag.y = ldsX[mrow * LDS_STRIDE + k0 + khalf + 1];
        const float* wp = W + (size_t)(k0 + khalf) * D_FEAT + n0 + mrow;
        bfrag.x = wp[0];        // K = k0+khalf
        bfrag.y = wp[D_FEAT];   // K = k0+khalf+1
        acc = __builtin_amdgcn_wmma_f32_16x16x4_f32(
            /*neg_a=*/false, afrag, /*neg_b=*/false, bfrag,
            /*c_mod=*/(short)0, acc, /*reuse_a=*/false, /*reuse_b=*/false);
    }

    // Store D tile: lane holds rows (lane>>4)*8 + r, column n0 + (lane&15).
    const int mtop = mbase + (lane >> 4) * 8;
    float* outp = support + (size_t)mtop * D_FEAT + n0 + (lane & 15);
    #pragma unroll
    for (int r = 0; r < 8; ++r) {
        if (mtop + r < n_nodes) outp[(size_t)r * D_FEAT] = acc[r];
    }
}

// ---------------------------------------------------------------------------
// Kernel 2: out[n][d] = bias[d]   (vectorized init so scatter is pure atomics)
// ---------------------------------------------------------------------------
__global__ void __launch_bounds__(256)
gcn_init_bias(const float* __restrict__ bias, float* __restrict__ out,
              size_t total) {
    size_t i = ((size_t)blockIdx.x * blockDim.x + threadIdx.x) * 4;
    if (i < total) {
        v4f b = *(const v4f*)(bias + (i & (D_FEAT - 1)));  // i%128, i is 4-aligned
        *(v4f*)(out + i) = b;
    }
}

// ---------------------------------------------------------------------------
// Kernel 3: COO scatter. One wave per edge (grid-stride): the 32 lanes gather
// support[c] as one coalesced 512B read (float4/lane) and scatter with four
// device-scope global_atomic_add_f32 per lane. support (51.2 MB) is L2-resident
// on MI455X (192 MB L2), so gather + atomic traffic stays on-die.
// ---------------------------------------------------------------------------
__global__ void __launch_bounds__(256)
gcn_spmm_scatter(const long long* __restrict__ erow,
                 const long long* __restrict__ ecol,
                 const float*     __restrict__ eval,
                 const float*     __restrict__ support,
                 float*           __restrict__ out,
                 int n_edges) {
    const int lane    = threadIdx.x & 31;
    const int wid     = (blockIdx.x * (blockDim.x >> 5)) + (threadIdx.x >> 5);
    const int n_waves = gridDim.x * (blockDim.x >> 5);

    for (int e = wid; e < n_edges; e += n_waves) {
        const int   r = (int)erow[e];
        const int   c = (int)ecol[e];
        const float v = eval[e];

        const v4f s = *(const v4f*)(support + (size_t)c * D_FEAT + lane * 4);
        float* dst  = out + (size_t)r * D_FEAT + lane * 4;
        global_fadd_dev(dst + 0, v * s.x);
        global_fadd_dev(dst + 1, v * s.y);
        global_fadd_dev(dst + 2, v * s.z);
        global_fadd_dev(dst + 3, v * s.w);
    }
}

// ---------------------------------------------------------------------------
extern "C" void kernel_launch(void* const* d_in, const int* in_sizes, int n_in,
                              void* d_out, int out_size, void* d_ws, size_t ws_size,
                              hipStream_t stream) {
    const float*     X    = (const float*)d_in[0];       // [N, 128] fp32
    const float*     W    = (const float*)d_in[1];       // [128, 128] fp32
    const float*     bias = (const float*)d_in[2];       // [128] fp32
    const long long* erow = (const long long*)d_in[3];   // [E] int64
    const long long* ecol = (const long long*)d_in[4];   // [E] int64
    const float*     eval = (const float*)d_in[5];       // [E] fp32

    float* out     = (float*)d_out;
    float* support = (float*)d_ws;                       // [N, 128] fp32 = 51.2 MB

    const int n_nodes = in_sizes[0] / D_FEAT;
    const int n_edges = in_sizes[3];
    const size_t total_out = (size_t)n_nodes * D_FEAT;

    // 1) support = X @ W   (fp32 WMMA)
    const int mblocks = (n_nodes + 15) / 16;
    gcn_gemm_f32_wmma<<<mblocks, 256, 0, stream>>>(X, W, support, n_nodes);

    // 2) out = bias (broadcast)
    const int ithreads = (int)((total_out / 4 + 255) / 256);
    gcn_init_bias<<<ithreads, 256, 0, stream>>>(bias, out, total_out);

    // 3) out[r] += val * support[c]  (atomic scatter)
    int blocks = (n_edges + 7) / 8;          // 8 waves per block
    if (blocks > 4096) blocks = 4096;        // persistent waves, grid-stride
    gcn_spmm_scatter<<<blocks, 256, 0, stream>>>(erow, ecol, eval, support,
                                                 out, n_edges);
}